// KeopsDFTD3Module_25744033972992
// MI455X (gfx1250) — compile-verified
//
#include <hip/hip_runtime.h>

// D3-BJ all-pairs dispersion energy, N=8192 atoms, single-scalar output.
//
// d2[i][j] = |p_i|^2 + |p_j|^2 - 2 p_i.p_j  ==  one V_WMMA_F32_16X16X4_F32:
//   A(16x4) row i = [x_i, y_i, z_i, 1]
//   B(4x16) col j = [-2x_j, -2y_j, -2z_j, |p_j|^2]
//   C(16x16)      = |p_i|^2 broadcast along rows
// Post-WMMA per element: -(s6*x8 + s8*x6) * rcp(x6*x8) with one NR step,
// where x6 = d2^3 + (a1+a2)^6 + eps, x8 = d2^4 + (a1+a2)^8 + eps.
// Diagonal excluded by exact index test (Gram form cancels on the diagonal).

typedef __attribute__((ext_vector_type(2))) float v2f;
typedef __attribute__((ext_vector_type(8))) float v8f;

#define N_ATOMS 8192
#define TILE 16
#define NT (N_ATOMS / TILE)        // 512 row/col tiles
#define BLOCK_THREADS 256          // 8 wave32 waves per block
#define WAVES_PER_BLOCK (BLOCK_THREADS / 32)
#define EPSF 1e-12f

__global__ __launch_bounds__(BLOCK_THREADS)
void d3_pair_tiles_kernel(const float* __restrict__ pos,
                          const float* __restrict__ a1p,
                          const float* __restrict__ a2p,
                          const float* __restrict__ s6p,
                          const float* __restrict__ s8p,
                          float* __restrict__ block_partials)
{
    const int lane = threadIdx.x & 31;
    const int wave = threadIdx.x >> 5;
    const int hi   = lane >> 4;      // 0: lanes 0-15, 1: lanes 16-31
    const int lo   = lane & 15;

    const float a1 = a1p[0], a2 = a2p[0];
    const float s6 = s6p[0], s8 = s8p[0];
    const float t  = a1 + a2;
    const float t2 = t * t;
    const float t6 = t2 * t2 * t2;
    const float t8 = t6 * t2;
    const float den6c = t6 + EPSF;   // added to d6
    const float den8c = t8 + EPSF;   // added to d8

    // ---- Row tile (fixed for this block/wave) ----
    const int i0   = blockIdx.x * TILE;
    const int irow = i0 + lo;                       // this lane's A row
    const float xi = pos[3 * irow + 0];
    const float yi = pos[3 * irow + 1];
    const float zi = pos[3 * irow + 2];
    const float qi = fmaf(xi, xi, fmaf(yi, yi, zi * zi));   // |p_i|^2

    // A operand layout (f32 16x4): v0 = K0 (lanes 0-15) / K2 (lanes 16-31),
    //                              v1 = K1 (lanes 0-15) / K3 (lanes 16-31)
    v2f A;
    A.x = hi ? zi : xi;        // K2=z | K0=x
    A.y = hi ? 1.0f : yi;      // K3=1 | K1=y

    // C init: c[v] holds row M = v + 8*hi -> broadcast |p_row|^2 from lane M
    v8f Cinit;
#pragma unroll
    for (int v = 0; v < 8; ++v) {
        const int src = (v + (hi << 3)) << 2;       // byte lane address
        Cinit[v] = __int_as_float(
            __builtin_amdgcn_ds_bpermute(src, __float_as_int(qi)));
    }

    // ---- Sweep column tiles ----
    float acc = 0.0f;
    for (int jt = wave; jt < NT; jt += WAVES_PER_BLOCK) {
        const int j0   = jt * TILE;
        const int jcol = j0 + lo;                   // this lane's B column
        const float xj = pos[3 * jcol + 0];
        const float yj = pos[3 * jcol + 1];
        const float zj = pos[3 * jcol + 2];
        const float qj = fmaf(xj, xj, fmaf(yj, yj, zj * zj));

        v2f B;
        B.x = hi ? (-2.0f * zj) : (-2.0f * xj);     // K2=-2z | K0=-2x
        B.y = hi ? qj : (-2.0f * yj);               // K3=qj  | K1=-2y

        // d2 tile = A*B + C  (one matrix-core op per 256 pairs)
        v8f D = __builtin_amdgcn_wmma_f32_16x16x4_f32(
            false, A, false, B, (short)0, Cinit, false, false);

#pragma unroll
        for (int v = 0; v < 8; ++v) {
            const float d2 = D[v];
            const float d6 = d2 * d2 * d2;
            const float d8 = d6 * d2;
            const float x6 = d6 + den6c;
            const float x8 = d8 + den8c;
            // -(s6/x6 + s8/x8) = -(s6*x8 + s8*x6) / (x6*x8): single rcp + NR
            const float prod = x6 * x8;             // < 2e28, no fp32 overflow
            const float num  = fmaf(s6, x8, s8 * x6);
            float r = __builtin_amdgcn_rcpf(prod);
            r = r * (2.0f - prod * r);              // Newton refine
            const float term = -num * r;
            const int gi = i0 + v + (hi << 3);      // global row of element v
            acc += (gi != jcol) ? term : 0.0f;      // exact diagonal mask
        }
    }

    // ---- Deterministic block reduction ----
    __shared__ float sred[BLOCK_THREADS];
    sred[threadIdx.x] = acc;
    __syncthreads();
#pragma unroll
    for (int s = BLOCK_THREADS / 2; s > 0; s >>= 1) {
        if (threadIdx.x < (unsigned)s)
            sred[threadIdx.x] += sred[threadIdx.x + s];
        __syncthreads();
    }
    if (threadIdx.x == 0)
        block_partials[blockIdx.x] = sred[0];
}

__global__ __launch_bounds__(NT)
void d3_final_reduce_kernel(const float* __restrict__ partials,
                            float* __restrict__ out)
{
    __shared__ float sred[NT];
    sred[threadIdx.x] = partials[threadIdx.x];
    __syncthreads();
#pragma unroll
    for (int s = NT / 2; s > 0; s >>= 1) {
        if (threadIdx.x < (unsigned)s)
            sred[threadIdx.x] += sred[threadIdx.x + s];
        __syncthreads();
    }
    if (threadIdx.x == 0)
        out[0] = sred[0];
}

extern "C" void kernel_launch(void* const* d_in, const int* in_sizes, int n_in,
                              void* d_out, int out_size, void* d_ws, size_t ws_size,
                              hipStream_t stream)
{
    // setup_inputs() order:
    // [0] atomic_numbers (int, N)   -- dead (multiplied by 0.0 in reference)
    // [1] positions      (float, N*3)
    // [2] r2r4           (float, 95) -- dead
    // [3] a1  [4] a2  [5] s6  [6] s8  (float scalars)
    const float* pos = (const float*)d_in[1];
    const float* a1  = (const float*)d_in[3];
    const float* a2  = (const float*)d_in[4];
    const float* s6  = (const float*)d_in[5];
    const float* s8  = (const float*)d_in[6];

    float* partials = (float*)d_ws;               // NT floats = 2 KB scratch

    d3_pair_tiles_kernel<<<NT, BLOCK_THREADS, 0, stream>>>(
        pos, a1, a2, s6, s8, partials);
    d3_final_reduce_kernel<<<1, NT, 0, stream>>>(partials, (float*)d_out);
}